// WatchReadLookupLoss_1133871366521
// MI455X (gfx1250) — compile-verified
//
#include <hip/hip_runtime.h>
#include <math.h>

typedef __attribute__((ext_vector_type(2))) float v2f;
typedef __attribute__((ext_vector_type(8))) float v8f;

#define INV_TEMP 14.2857142857142857f   // 1 / 0.07; also the lse shift (max possible dist)
#define KC 64                            // K-chunk staged in LDS per iteration

// Issue one 16-byte async global->LDS copy (gfx1250, tracked by ASYNCcnt).
// LDS address = low 32 bits of the generic shared pointer (ISA: flat->LDS uses addr[31:0]).
__device__ __forceinline__ void async_copy_b128(const float* gptr, const float* lds_ptr)
{
    unsigned lds_off = (unsigned)(unsigned long long)lds_ptr;
    asm volatile("global_load_async_to_lds_b128 %0, %1, off"
                 :: "v"(lds_off), "v"(gptr)
                 : "memory");
}

__device__ __forceinline__ void wait_async_all()
{
    asm volatile("s_wait_asynccnt 0" ::: "memory");
}

// ---------------------------------------------------------------------------
// Kernel 1: metadata + accumulator init (single block)
// ---------------------------------------------------------------------------
__global__ void meta_kernel(const int* __restrict__ targets_dict,
                            const int* __restrict__ batch_dict,
                            const int* __restrict__ is_mouth,
                            const int* __restrict__ uniq,
                            const int* __restrict__ nb_ptr,
                            int Nb, int Nd, int G,
                            float* numS, float* t1S, float* t3S, float* rowA,
                            int* gidx, int* mouth, int* gbatch, int* gmouth)
{
    const int tid = threadIdx.x;
    const int T   = blockDim.x;
    const int B   = nb_ptr[0];

    for (int g = tid; g < G; g += T) { numS[g] = 0.f; t1S[g] = 0.f; t3S[g] = 0.f; }
    for (int i = tid; i < Nb; i += T) rowA[i] = 0.f;

    // g_idx[j] = searchsorted(unique_targets, targets_dict[j])  (lower_bound)
    for (int j = tid; j < Nd; j += T) {
        int t = targets_dict[j];
        int lo = 0, hi = G;
        while (lo < hi) { int mid = (lo + hi) >> 1; if (uniq[mid] < t) lo = mid + 1; else hi = mid; }
        gidx[j] = lo;
    }
    __syncthreads();

    // mouth_word[b] = segment_max over batch of where(is_mouth==1, targets_dict, -1)
    for (int b = tid; b < B; b += T) {
        int m = -1;
        for (int j = 0; j < Nd; ++j)
            if (batch_dict[j] == b && is_mouth[j] == 1) m = max(m, targets_dict[j]);
        mouth[b] = m;
    }
    // group_batch / group_is_mouth = segment_max over g_idx
    for (int g = tid; g < G; g += T) {
        int gb = -2147483647, gm = -2147483647;
        for (int j = 0; j < Nd; ++j)
            if (gidx[j] == g) { gb = max(gb, batch_dict[j]); gm = max(gm, is_mouth[j]); }
        gbatch[g] = gb; gmouth[g] = gm;
    }
}

// ---------------------------------------------------------------------------
// Kernel 2: L2-normalize rows of fb and fd (one wave32 per row)
// ---------------------------------------------------------------------------
__global__ void norm_kernel(const float* __restrict__ fb, const float* __restrict__ fd,
                            float* __restrict__ fbN, float* __restrict__ fdN,
                            int Nb, int Nd, int D)
{
    const int wave = threadIdx.x >> 5;
    const int lane = threadIdx.x & 31;
    const int r = blockIdx.x * (blockDim.x >> 5) + wave;
    if (r >= Nb + Nd) return;
    const float* src = (r < Nb) ? (fb + (size_t)r * D) : (fd + (size_t)(r - Nb) * D);
    float*       dst = (r < Nb) ? (fbN + (size_t)r * D) : (fdN + (size_t)(r - Nb) * D);
    float ss = 0.f;
    for (int k = lane; k < D; k += 32) { float v = src[k]; ss += v * v; }
    for (int m = 16; m >= 1; m >>= 1) ss += __shfl_xor(ss, m, 32);
    float inv = rsqrtf(ss);
    for (int k = lane; k < D; k += 32) dst[k] = src[k] * inv;
}

// ---------------------------------------------------------------------------
// Kernel 3: fused WMMA GEMM + exp/mask epilogue + segment reductions.
// Block = 8 waves; wave w owns the 16x16 tile (rowTile, blockIdx.y*8 + w).
// K is chunked: each chunk is staged into LDS with async global->LDS b128
// copies (ASYNCcnt), then consumed via ds_load + V_WMMA_F32_16X16X4_F32.
// ---------------------------------------------------------------------------
__global__ __launch_bounds__(256)
void gemm_kernel(const float* __restrict__ fbN, const float* __restrict__ fdN,
                 const int* __restrict__ targets_bsl1k,
                 const int* __restrict__ batch_bsl1k,
                 const int* __restrict__ batch_dict,
                 const int* __restrict__ uniq,
                 const int* __restrict__ gidx,
                 const int* __restrict__ mouth,
                 const int* __restrict__ gbatch,
                 const int* __restrict__ gmouth,
                 float* numS, float* t1S, float* t3S, float* rowA,
                 int Nb, int Nd, int D)
{
    __shared__ float sA[16 * KC];    //  4 KB: A tile, 16 rows x KC
    __shared__ float sB[128 * KC];   // 32 KB: 8 B tiles, 128 cols x KC

    const int tid  = threadIdx.x;
    const int wave = tid >> 5;
    const int lane = tid & 31;
    const int half = lane >> 4;      // 0: lanes 0-15, 1: lanes 16-31
    const int l15  = lane & 15;

    const int rowTile = blockIdx.x;
    const int row0    = rowTile << 4;
    const int colBase = blockIdx.y << 7;          // 128 columns per block
    const int col0    = colBase + (wave << 4);    // this wave's 16 columns

    // staging assignments (b128 = 4 floats per copy)
    const int aRow = tid >> 4;       // 0..15
    const int aSeg = (tid & 15) * 4; // 0..60

    v8f acc = {};
    for (int k0 = 0; k0 < D; k0 += KC) {
        // ---- async stage A (16 x KC) : one b128 per thread ----
        async_copy_b128(fbN + (size_t)(row0 + aRow) * D + k0 + aSeg,
                        &sA[aRow * KC + aSeg]);
        // ---- async stage B (128 x KC) : eight b128 per thread ----
        #pragma unroll
        for (int t = 0; t < 8; ++t) {
            int idx  = tid + t * 256;
            int bRow = idx >> 4;            // 0..127
            int bSeg = (idx & 15) * 4;      // 0..60
            async_copy_b128(fdN + (size_t)(colBase + bRow) * D + k0 + bSeg,
                            &sB[bRow * KC + bSeg]);
        }
        wait_async_all();     // each wave drains its own ASYNCcnt before the barrier
        __syncthreads();

        // ---- consume chunk: KC/4 WMMA steps from LDS ----
        const float* la = &sA[l15 * KC + 2 * half];
        const float* lb = &sB[(wave * 16 + l15) * KC + 2 * half];
        #pragma unroll
        for (int k = 0; k < KC; k += 4) {
            v2f a, b;
            a.x = la[k]; a.y = la[k + 1];
            b.x = lb[k]; b.y = lb[k + 1];
            acc = __builtin_amdgcn_wmma_f32_16x16x4_f32(
                /*neg_a=*/false, a, /*neg_b=*/false, b,
                /*c_mod=*/(short)0, acc, /*reuse_a=*/false, /*reuse_b=*/false);
        }
        __syncthreads();      // protect LDS before next chunk overwrites it
    }

    // Epilogue. C layout: VGPR v, lane -> (row = row0 + half*8 + v, col = col0 + l15)
    const int j  = col0 + l15;
    const int g  = gidx[j];
    const int bd = batch_dict[j];
    const int gb = gbatch[g], gm = gmouth[g], ut = uniq[g];

    float se_m = 0.f, sa = 0.f, sa_m = 0.f;
    float ra[8];
    const int rbase = row0 + half * 8;
    #pragma unroll
    for (int v = 0; v < 8; ++v) {
        const int i  = rbase + v;
        const int tb = targets_bsl1k[i];
        const int bb = batch_bsl1k[i];
        const float e = __expf(acc[v] * INV_TEMP - INV_TEMP); // exp(dist - c), c = 1/T
        const bool  m = (tb != -1) ? (ut == mouth[bb]) : ((gb == bb) && (gm == 0));
        const float a = (m && (bb != bd)) ? 0.f : e;          // A = where(DB, 0, E)
        if (m) { se_m += e; sa_m += a; }
        sa   += a;
        ra[v] = a;
    }

    // Column (= group) partial sums: lanes L and L+16 hold the same column.
    float c_num = se_m + __shfl_xor(se_m, 16, 32);
    float c_t1  = sa   + __shfl_xor(sa,   16, 32);
    float c_t3  = sa_m + __shfl_xor(sa_m, 16, 32);
    if (half == 0) {
        atomicAdd(&numS[g], c_num);
        atomicAdd(&t1S[g],  c_t1);
        atomicAdd(&t3S[g],  c_t3);
    }

    // Row partial sums: 16 lanes within a half hold the 16 columns of one row.
    #pragma unroll
    for (int v = 0; v < 8; ++v) {
        float r = ra[v];
        r += __shfl_xor(r, 1, 32);
        r += __shfl_xor(r, 2, 32);
        r += __shfl_xor(r, 4, 32);
        r += __shfl_xor(r, 8, 32);
        if (l15 == 0) atomicAdd(&rowA[rbase + v], r);
    }
}

// ---------------------------------------------------------------------------
// Kernel 4: t2[g], den/num -> mean(log(den) - log(num)) (single block)
// ---------------------------------------------------------------------------
__global__ void final_kernel(const int* __restrict__ targets_bsl1k,
                             const int* __restrict__ batch_bsl1k,
                             const int* __restrict__ uniq,
                             const int* __restrict__ mouth,
                             const int* __restrict__ gbatch,
                             const int* __restrict__ gmouth,
                             const float* __restrict__ numS,
                             const float* __restrict__ t1S,
                             const float* __restrict__ t3S,
                             const float* __restrict__ rowA,
                             int Nb, int G, float* out)
{
    __shared__ float red[256];
    const int tid = threadIdx.x;
    float acc = 0.f;
    for (int g = tid; g < G; g += blockDim.x) {
        const int gb = gbatch[g], gm = gmouth[g], ut = uniq[g];
        float t2 = 0.f;
        for (int i = 0; i < Nb; ++i) {
            const int tb = targets_bsl1k[i];
            const int bb = batch_bsl1k[i];
            const bool m = (tb != -1) ? (ut == mouth[bb]) : ((gb == bb) && (gm == 0));
            if (m) t2 += rowA[i];
        }
        const float den = t1S[g] + t2 - t3S[g];
        acc += logf(den) - logf(numS[g]);
    }
    red[tid] = acc;
    __syncthreads();
    for (int s = blockDim.x >> 1; s > 0; s >>= 1) {
        if (tid < s) red[tid] += red[tid + s];
        __syncthreads();
    }
    if (tid == 0) out[0] = red[0] / (float)G;
}

// ---------------------------------------------------------------------------
extern "C" void kernel_launch(void* const* d_in, const int* in_sizes, int n_in,
                              void* d_out, int out_size, void* d_ws, size_t ws_size,
                              hipStream_t stream)
{
    const float* fb            = (const float*)d_in[0];
    const float* fd            = (const float*)d_in[1];
    const int* targets_bsl1k   = (const int*)d_in[2];
    const int* targets_dict    = (const int*)d_in[3];
    const int* batch_bsl1k     = (const int*)d_in[4];
    const int* batch_dict      = (const int*)d_in[5];
    const int* is_mouth        = (const int*)d_in[6];
    const int* uniq            = (const int*)d_in[7];
    const int* nb_ptr          = (const int*)d_in[8];
    float* out                 = (float*)d_out;

    const int Nb = in_sizes[2];              // 4096
    const int Nd = in_sizes[3];              // 1024
    const int D  = in_sizes[0] / Nb;         // 256
    const int G  = in_sizes[7];              // 256

    // Workspace layout
    float* ws    = (float*)d_ws;
    float* fbN   = ws;                            // Nb*D
    float* fdN   = fbN + (size_t)Nb * D;          // Nd*D
    float* numS  = fdN + (size_t)Nd * D;          // G
    float* t1S   = numS + G;                      // G
    float* t3S   = t1S + G;                       // G
    float* rowA  = t3S + G;                       // Nb
    int*   gidx   = (int*)(rowA + Nb);            // Nd
    int*   mouth  = gidx + Nd;                    // B (<= G since G = B*K)
    int*   gbatch = mouth + G;                    // G
    int*   gmouth = gbatch + G;                   // G

    meta_kernel<<<1, 256, 0, stream>>>(targets_dict, batch_dict, is_mouth, uniq, nb_ptr,
                                       Nb, Nd, G, numS, t1S, t3S, rowA,
                                       gidx, mouth, gbatch, gmouth);

    const int rowsPerBlock = 8; // 8 waves of 32
    const int nrmBlocks = (Nb + Nd + rowsPerBlock - 1) / rowsPerBlock;
    norm_kernel<<<nrmBlocks, 256, 0, stream>>>(fb, fd, fbN, fdN, Nb, Nd, D);

    dim3 ggrid(Nb / 16, Nd / 128);  // exact: Nb mult of 16, Nd mult of 128
    gemm_kernel<<<ggrid, 256, 0, stream>>>(fbN, fdN, targets_bsl1k, batch_bsl1k, batch_dict,
                                           uniq, gidx, mouth, gbatch, gmouth,
                                           numS, t1S, t3S, rowA, Nb, Nd, D);

    final_kernel<<<1, 256, 0, stream>>>(targets_bsl1k, batch_bsl1k, uniq, mouth,
                                        gbatch, gmouth, numS, t1S, t3S, rowA,
                                        Nb, G, out);
}